// dc_9019431322178
// MI455X (gfx1250) — compile-verified
//
#include <hip/hip_runtime.h>
#include <math.h>

typedef __attribute__((ext_vector_type(2))) float v2f;
typedef __attribute__((ext_vector_type(8))) float v8f;

#define NIMG 4
#define NN 256
#define NANG 180
#define IMG (NN * NN)            // 65536
#define IMGS (NIMG * IMG)        // 262144
#define SGM (NANG * NN)          // 46080
#define SGMS (NIMG * SGM)        // 184320

// ---------------------------------------------------------------------------
// Prep kernel: trig table + analytic ramp-filter spatial kernel table.
//   trig[2a]=cos(theta_a), trig[2a+1]=sin(theta_a) (theta given in degrees)
//   gtab[d] = 0.5 (d==0), -2/(pi d)^2 (d odd), 0 (d even>0)
// gtab is the exact spatial-domain equivalent of the reference's FFT ramp
// filter, since ifft(2*Re(fft(f))) = f + f_reversed (closed form).
// ---------------------------------------------------------------------------
__global__ void prep_kernel(const float* __restrict__ theta,
                            float* __restrict__ trig,
                            float* __restrict__ gtab) {
    int t = threadIdx.x;     // 0..255
    if (t < NANG) {
        float rad = theta[t] * 0.017453292519943295f;
        trig[2 * t]     = cosf(rad);
        trig[2 * t + 1] = sinf(rad);
    }
    float fd = (float)t;
    float oddv = -0.20264236728467555f / (fd * fd);   // -(2/pi^2) / d^2
    gtab[t] = (t == 0) ? 0.5f : ((t & 1) ? oddv : 0.0f);
}

// ---------------------------------------------------------------------------
// WMMA ramp-filter GEMM:  filtT[a][m] = sum_k gtab[|m-k|] * sino(k, a)
// sino element (k, n) at S[k*sk + n*sn]  (sk=NANG,sn=1 for the input b layout
// [m][a]; sk=1,sn=NN for the transposed sinogram produced by radon_kernel).
// One wave (32 threads) computes one 16x16 output tile via 64 chained
// V_WMMA_F32_16X16X4_F32 ops; A-matrix values come from the 1KB gtab table
// (no divisions, no divergence -> EXEC stays all-ones across every WMMA).
// ---------------------------------------------------------------------------
__global__ void filt_gemm_kernel(const float* __restrict__ sino,
                                 float* __restrict__ filtT,
                                 const float* __restrict__ gtab,
                                 int sk, int sn) {
    const int lane = threadIdx.x;        // 0..31
    const int half = lane >> 4;          // 0 or 1
    const int l16  = lane & 15;
    const int mb   = blockIdx.x * 16;    // detector-bin tile (0..240)
    const int nb   = blockIdx.y * 16;    // angle tile (0..176)
    const int img  = blockIdx.z;

    const float* S  = sino  + img * SGM;
    float*       Fo = filtT + img * SGM;

    const int n    = nb + l16;                       // angle this lane owns in B
    const int nl   = n < NANG ? n : (NANG - 1);      // clamped for safe loads
    const int mrow = mb + l16;                       // output row this lane owns in A

    v8f acc = {};
    for (int k0 = 0; k0 < NN; k0 += 4) {
        const int ka = k0 + 2 * half;                // K for VGPR0 of this lane
        int d0 = mrow - ka;      d0 = d0 < 0 ? -d0 : d0;
        int d1 = mrow - ka - 1;  d1 = d1 < 0 ? -d1 : d1;
        v2f av, bv;
        // A layout (16x4 f32): lane=M, VGPR0 holds K=2*half, VGPR1 K=2*half+1
        av.x = gtab[d0];
        av.y = gtab[d1];
        // B layout (4x16 f32): lane%16=N, VGPR0 holds K=2*half, VGPR1 K=2*half+1
        bv.x = S[ka * sk + nl * sn];
        bv.y = S[(ka + 1) * sk + nl * sn];
        acc = __builtin_amdgcn_wmma_f32_16x16x4_f32(
            /*neg_a=*/false, av, /*neg_b=*/false, bv,
            /*c_mod=*/(short)0, acc, /*reuse_a=*/false, /*reuse_b=*/false);
    }

    if (n < NANG) {
        // C/D layout: VGPR v -> M = mb + v + 8*half, N = n
        #pragma unroll
        for (int v = 0; v < 8; ++v) {
            int m = mb + v + half * 8;
            Fo[n * NN + m] = acc[v];
        }
    }
}

// ---------------------------------------------------------------------------
// Mask for radon (circle=True in _radon): pm = p * ((y-c)^2+(x-c)^2 <= c^2),
// c = 127.5.
// ---------------------------------------------------------------------------
__global__ void mask_kernel(const float* __restrict__ p, float* __restrict__ pm) {
    int i = blockIdx.x * blockDim.x + threadIdx.x;   // 0..IMGS-1
    int rem = i & (IMG - 1);
    float u = (float)(rem >> 8) - 127.5f;
    float v = (float)(rem & 255) - 127.5f;
    pm[i] = (u * u + v * v <= 16256.25f) ? p[i] : 0.0f;
}

// ---------------------------------------------------------------------------
// Radon transform: sinoT[a][r] = sum_y bilinear(pm; sr, sc) with
// sr = u*ca - v*sa + c, sc = u*sa + v*ca + c, u=y-c, v=r-c, zero outside
// (map_coordinates order=1, mode='constant', cval=0).
// One block per (angle, image); thread = detector bin r.
// ---------------------------------------------------------------------------
__global__ void radon_kernel(const float* __restrict__ pm,
                             float* __restrict__ sinoT,
                             const float* __restrict__ trig) {
    const int a   = blockIdx.x;
    const int img = blockIdx.y;
    const int r   = threadIdx.x;
    const float ca = trig[2 * a], sa = trig[2 * a + 1];
    const float c = 127.5f;
    const float* im = pm + img * IMG;
    const float v = (float)r - c;
    float acc = 0.0f;
    for (int y = 0; y < NN; ++y) {
        float u  = (float)y - c;
        float sr = u * ca - v * sa + c;
        float sc = u * sa + v * ca + c;
        float fr = floorf(sr), fc = floorf(sc);
        int i0 = (int)fr, j0 = (int)fc;
        float wr = sr - fr, wc = sc - fc;
        bool r0 = (unsigned)i0 < 256u, r1 = (unsigned)(i0 + 1) < 256u;
        bool c0 = (unsigned)j0 < 256u, c1 = (unsigned)(j0 + 1) < 256u;
        float v00 = (r0 && c0) ? im[i0 * NN + j0]           : 0.0f;
        float v01 = (r0 && c1) ? im[i0 * NN + j0 + 1]       : 0.0f;
        float v10 = (r1 && c0) ? im[(i0 + 1) * NN + j0]     : 0.0f;
        float v11 = (r1 && c1) ? im[(i0 + 1) * NN + j0 + 1] : 0.0f;
        acc += (1.0f - wr) * ((1.0f - wc) * v00 + wc * v01)
             +         wr  * ((1.0f - wc) * v10 + wc * v11);
    }
    sinoT[img * SGM + a * NN + r] = acc;
}

// ---------------------------------------------------------------------------
// Filtered backprojection + lam*add:
// dest = circle * (pi/(2A)) * sum_a interp(filtT[a], t) + lam*addsrc,
// t = (j-128)*cos - (i-128)*sin, inside t in [-128,127], circle radius 128.
// ---------------------------------------------------------------------------
__global__ void backproj_kernel(const float* __restrict__ filtT,
                                const float* __restrict__ addsrc,
                                const float* __restrict__ lam,
                                float* __restrict__ dest,
                                const float* __restrict__ trig) {
    const int i   = blockIdx.x;
    const int img = blockIdx.y;
    const int j   = threadIdx.x;
    const float* F = filtT + img * SGM;
    const float xpr = (float)(i - 128);
    const float ypr = (float)(j - 128);
    float acc = 0.0f;
    for (int a = 0; a < NANG; ++a) {
        float ca = trig[2 * a], sa = trig[2 * a + 1];
        float t = ypr * ca - xpr * sa;
        if (t >= -128.0f && t <= 127.0f) {
            float pos = t + 128.0f;
            float fk = floorf(pos);
            int k = (int)fk;
            float w = pos - fk;
            int k1 = k < (NN - 1) ? k + 1 : (NN - 1);
            float f0 = F[a * NN + k];
            acc += f0 + w * (F[a * NN + k1] - f0);
        }
    }
    bool inc = (xpr * xpr + ypr * ypr) <= 16384.0f;
    float outv = inc ? acc * 0.008726646259971648f : 0.0f;   // pi/(2*180)
    size_t idx = (size_t)img * IMG + i * NN + j;
    dest[idx] = outv + lam[0] * addsrc[idx];
}

// ---------------------------------------------------------------------------
// CG init: p = r; x = 0; rsold = r*r   (r already holds rhs)
// ---------------------------------------------------------------------------
__global__ void init_kernel(const float* __restrict__ r, float* __restrict__ p,
                            float* __restrict__ x, float* __restrict__ rs0) {
    int i = blockIdx.x * blockDim.x + threadIdx.x;
    float rv = r[i];
    p[i] = rv;
    rs0[i] = rv * rv;
    x[i] = 0.0f;
}

// alpha = sqrt(sum rsold^2) / sqrt(sum (p*Ap)^2); also stash sqrt(sum rsold^2)
__global__ void alpha_kernel(const float* __restrict__ rs0,
                             const float* __restrict__ p,
                             const float* __restrict__ Ap,
                             float* __restrict__ scal) {
    __shared__ float s1[256];
    __shared__ float s2[256];
    const int img = blockIdx.x;
    const int t = threadIdx.x;
    const float* R = rs0 + img * IMG;
    const float* P = p   + img * IMG;
    const float* Q = Ap  + img * IMG;
    float a1 = 0.0f, a2 = 0.0f;
    for (int i = t; i < IMG; i += 256) {
        float rv = R[i];  a1 += rv * rv;
        float pq = P[i] * Q[i];  a2 += pq * pq;
    }
    s1[t] = a1; s2[t] = a2; __syncthreads();
    for (int s = 128; s > 0; s >>= 1) {
        if (t < s) { s1[t] += s1[t + s]; s2[t] += s2[t + s]; }
        __syncthreads();
    }
    if (t == 0) {
        float n1 = sqrtf(s1[0]);
        scal[img]     = n1 / sqrtf(s2[0]);   // alpha
        scal[8 + img] = n1;                  // ||rsold||
    }
}

// x += alpha*p; r -= alpha*Ap; rsnew = r*r
__global__ void update1_kernel(float* __restrict__ x, float* __restrict__ r,
                               const float* __restrict__ p, const float* __restrict__ Ap,
                               float* __restrict__ rs1, const float* __restrict__ scal) {
    int i = blockIdx.x * blockDim.x + threadIdx.x;
    float al = scal[i >> 16];
    x[i] += al * p[i];
    float rv = r[i] - al * Ap[i];
    r[i] = rv;
    rs1[i] = rv * rv;
}

// beta = sqrt(sum rsnew^2) / ||rsold||
__global__ void beta_kernel(const float* __restrict__ rs1, float* __restrict__ scal) {
    __shared__ float s1[256];
    const int img = blockIdx.x;
    const int t = threadIdx.x;
    const float* R = rs1 + img * IMG;
    float a1 = 0.0f;
    for (int i = t; i < IMG; i += 256) { float rv = R[i]; a1 += rv * rv; }
    s1[t] = a1; __syncthreads();
    for (int s = 128; s > 0; s >>= 1) {
        if (t < s) s1[t] += s1[t + s];
        __syncthreads();
    }
    if (t == 0) scal[4 + img] = sqrtf(s1[0]) / scal[8 + img];
}

// p = r + beta*p
__global__ void update2_kernel(float* __restrict__ p, const float* __restrict__ r,
                               const float* __restrict__ scal) {
    int i = blockIdx.x * blockDim.x + threadIdx.x;
    float be = scal[4 + (i >> 16)];
    p[i] = r[i] + be * p[i];
}

__global__ void copy_kernel(const float* __restrict__ src, float* __restrict__ dst) {
    int i = blockIdx.x * blockDim.x + threadIdx.x;
    dst[i] = src[i];
}

// ---------------------------------------------------------------------------
extern "C" void kernel_launch(void* const* d_in, const int* in_sizes, int n_in,
                              void* d_out, int out_size, void* d_ws, size_t ws_size,
                              hipStream_t stream) {
    const float* zk    = (const float*)d_in[0];  // [4,1,256,256]
    const float* bsin  = (const float*)d_in[1];  // [4,1,256,180]
    const float* lam   = (const float*)d_in[2];  // [1]
    const float* theta = (const float*)d_in[3];  // [180]
    float* out = (float*)d_out;

    float* w   = (float*)d_ws;
    float* X   = w;
    float* R   = w + 1 * IMGS;
    float* P   = w + 2 * IMGS;
    float* AP  = w + 3 * IMGS;
    float* PM  = w + 4 * IMGS;
    float* RS0 = w + 5 * IMGS;
    float* RS1 = w + 6 * IMGS;
    float* SN  = w + 7 * IMGS;            // sinogram (transposed layout)
    float* FT  = SN + SGMS;               // filtered sinogram (transposed)
    float* TR  = FT + SGMS;               // trig table (360)
    float* SC  = TR + 384;                // scalars: alpha[4], beta[4], nrm[4]
    float* GT  = SC + 16;                 // ramp-filter table (256)

    dim3 gGemm(16, 12, NIMG);             // 16 m-tiles x ceil(180/16) n-tiles
    dim3 gBp(NN, NIMG);
    dim3 gRad(NANG, NIMG);
    const int EB = IMGS / 256;

    // rhs = iradon(b) + lam*z  -> stored in R
    prep_kernel<<<1, 256, 0, stream>>>(theta, TR, GT);
    filt_gemm_kernel<<<gGemm, 32, 0, stream>>>(bsin, FT, GT, NANG, 1);
    backproj_kernel<<<gBp, NN, 0, stream>>>(FT, zk, lam, R, TR);
    init_kernel<<<EB, 256, 0, stream>>>(R, P, X, RS0);

    float* rsA = RS0;
    float* rsB = RS1;
    for (int it = 0; it < 5; ++it) {
        // Ap = iradon(radon(p)) + lam*p
        mask_kernel<<<EB, 256, 0, stream>>>(P, PM);
        radon_kernel<<<gRad, NN, 0, stream>>>(PM, SN, TR);
        filt_gemm_kernel<<<gGemm, 32, 0, stream>>>(SN, FT, GT, 1, NN);
        backproj_kernel<<<gBp, NN, 0, stream>>>(FT, P, lam, AP, TR);
        // CG scalar + vector updates
        alpha_kernel<<<NIMG, 256, 0, stream>>>(rsA, P, AP, SC);
        update1_kernel<<<EB, 256, 0, stream>>>(X, R, P, AP, rsB, SC);
        beta_kernel<<<NIMG, 256, 0, stream>>>(rsB, SC);
        update2_kernel<<<EB, 256, 0, stream>>>(P, R, SC);
        float* tmp = rsA; rsA = rsB; rsB = tmp;
    }

    copy_kernel<<<EB, 256, 0, stream>>>(X, out);
}